// GATv2_28252294873768
// MI455X (gfx1250) — compile-verified
//
#include <hip/hip_runtime.h>
#include <math.h>

// GATv2 (2 conv layers + linear + log_softmax) for MI455X / gfx1250.
// Dense GEMMs: V_WMMA_F32_16X16X4_F32, 2 M-tiles per wave sharing B fragments.
// A-tile staging: Tensor Data Mover (tensor_load_to_lds) when available.
// Edge pipeline: 3 passes with float atomics; node features are L2-resident.

typedef __attribute__((ext_vector_type(2))) float v2f;
typedef __attribute__((ext_vector_type(8))) float v8f;
typedef unsigned int u32x4 __attribute__((ext_vector_type(4)));
typedef int i32x8 __attribute__((ext_vector_type(8)));
typedef int i32x4 __attribute__((ext_vector_type(4)));

#define NN    10000
#define EE    320000
#define ETOT  (EE + NN)      // edges + self loops = 330000
#define DD    256
#define DOUT  64
#define SLOPE 0.2f

__device__ __forceinline__ float lrelu(float v) { return v > 0.0f ? v : SLOPE * v; }

// Float atomic max via monotone bit trick (init to -inf).
__device__ __forceinline__ void atomicMaxF(float* addr, float val) {
  if (val >= 0.0f)
    atomicMax((int*)addr, __float_as_int(val));
  else
    atomicMin((unsigned int*)addr, __float_as_uint(val));
}

#if __has_builtin(__builtin_amdgcn_tensor_load_to_lds)
#define USE_TDM 1
#else
#define USE_TDM 0
#endif

#if USE_TDM
// Issue one TDM descriptor moving a [rows<=32] x 256 f32 tile (row-major,
// stride 256) from global to LDS. tensor_dim1 = rowsRemaining makes the TDM
// zero-fill reads past the last row (tail handling for M % 32 != 0).
// This toolchain (clang-23 / therock) exposes the 6-arg builtin:
//   (u32x4 g0, i32x8 g1, i32x4 g2, i32x4 g3, i32x8 extra, i32 cpol)
__device__ __forceinline__ void tdm_load_tile(const float* gsrc,
                                              unsigned ldsByteOff,
                                              int rowsRemaining) {
  unsigned long long ga = (unsigned long long)(const void*)gsrc;
  u32x4 g0;
  g0[0] = 1u;                                             // count=1 (valid D#)
  g0[1] = ldsByteOff;                                     // lds_addr
  g0[2] = (unsigned)(ga & 0xFFFFFFFFu);                   // global_addr[31:0]
  g0[3] = (unsigned)((ga >> 32) & 0x1FFFFFFu) | (2u << 30); // addr[56:32]|type=2
  i32x8 g1;
  g1[0] = (int)(2u << 16);                 // wg_mask=0, data_size=2 (4 bytes)
  g1[1] = (int)(256u << 16);               // tensor_dim0[15:0]=256 (in [31:16])
  g1[2] = (int)((unsigned)rowsRemaining << 16); // dim0 hi=0 | tensor_dim1 lo16
  g1[3] = (int)(256u << 16);               // tensor_dim1 hi=0 | tile_dim0=256
  g1[4] = 32;                              // tile_dim1=32, tile_dim2=0
  g1[5] = 256;                             // tensor_dim0_stride[31:0]=256
  g1[6] = 0;                               // stride hi | tensor_dim1_stride lo
  g1[7] = 0;
  i32x4 z4 = {0, 0, 0, 0};                 // groups 2/3 unused (2D tile)
  i32x8 z8 = {0, 0, 0, 0, 0, 0, 0, 0};
  __builtin_amdgcn_tensor_load_to_lds(g0, g1, z4, z4, z8, 0);
}
__device__ __forceinline__ void tdm_wait() {
#if __has_builtin(__builtin_amdgcn_s_wait_tensorcnt)
  __builtin_amdgcn_s_wait_tensorcnt(0);
#else
  asm volatile("s_wait_tensorcnt 0x0" ::: "memory");
#endif
}
#endif

// ---------------------------------------------------------------------------
// Y[M x NC] = X[M x 256] @ W[256 x NC] + bias.
// Block stages a 32-row A stripe in LDS (TDM when available); each wave owns
// one 16-wide N tile and TWO 16-row M tiles so every B fragment feeds 2 WMMAs.
// WMMA f32 16x16x4: A lanes 0-15 hold M=0..15 K={k,k+1}, lanes 16-31 K={k+2,k+3};
// B mirrors with N across lanes; C/D: vgpr r holds M = r + 8*(lane>=16), N=lane&15.
// ---------------------------------------------------------------------------
template <int NC>
__global__ void __launch_bounds__(256) gemm_wmma(const float* __restrict__ X,
                                                 const float* __restrict__ W,
                                                 const float* __restrict__ bias,
                                                 float* __restrict__ Y,
                                                 int M) {
  __shared__ __align__(16) float As[32 * DD];   // 32 rows x 256 K = 32 KB
  const int tid  = threadIdx.x;
  const int wave = tid >> 5;
  const int lane = tid & 31;
  const int m0   = blockIdx.x * 32;

#if USE_TDM
  if (wave == 0) {
    unsigned ldsOff =
        (unsigned)(unsigned long long)(__attribute__((address_space(3))) float*)As;
    tdm_load_tile(X + (size_t)m0 * DD, ldsOff, M - m0);
    tdm_wait();
  }
#else
  {
    // Fallback: float4 staging; clamp source row so the tail never reads OOB
    // (garbage rows only affect C rows that are never stored).
    const float4* Xg  = (const float4*)X;
    float4*       Asv = (float4*)As;
    for (int i = tid; i < 32 * (DD / 4); i += blockDim.x) {
      int r  = i >> 6;                       // row in tile
      int c4 = i & 63;                       // float4 column
      int gr = m0 + r; if (gr > M - 1) gr = M - 1;
      Asv[i] = Xg[(size_t)gr * (DD / 4) + c4];
    }
  }
#endif
  __syncthreads();

  const int wavesPerBlock = blockDim.x >> 5;
  const int n0  = (blockIdx.y * wavesPerBlock + wave) * 16;
  const int r16 = lane & 15;            // M row (A) / N col (B,C)
  const int kh  = (lane >> 4) * 2;      // K sub-offset: 0 or 2

  v8f c0 = {}, c1 = {};
#pragma unroll 8
  for (int k0 = 0; k0 < DD; k0 += 4) {
    v2f a0, a1, b;
    a0.x = As[r16 * DD + k0 + kh];
    a0.y = As[r16 * DD + k0 + kh + 1];
    a1.x = As[(16 + r16) * DD + k0 + kh];
    a1.y = As[(16 + r16) * DD + k0 + kh + 1];
    b.x  = W[(size_t)(k0 + kh) * NC + n0 + r16];
    b.y  = W[(size_t)(k0 + kh + 1) * NC + n0 + r16];
    c0 = __builtin_amdgcn_wmma_f32_16x16x4_f32(false, a0, false, b, (short)0, c0,
                                               false, false);
    c1 = __builtin_amdgcn_wmma_f32_16x16x4_f32(false, a1, false, b, (short)0, c1,
                                               false, false);
  }

  const int   nn  = n0 + r16;
  const float bb  = bias[nn];
  const int   mb0 = m0 + 8 * (lane >> 4);
#pragma unroll
  for (int r = 0; r < 8; ++r) {
    int mm = mb0 + r;
    if (mm < M) Y[(size_t)mm * NC + nn] = c0[r] + bb;
  }
#pragma unroll
  for (int r = 0; r < 8; ++r) {
    int mm = mb0 + 16 + r;
    if (mm < M) Y[(size_t)mm * NC + nn] = c1[r] + bb;
  }
}

// ---------------------------------------------------------------------------
// Zero accumulators, -inf the per-node max, zero the per-node exp-sum.
// ---------------------------------------------------------------------------
__global__ void init_accum(float* __restrict__ acc, float* __restrict__ mx,
                           float* __restrict__ sm) {
  int i = blockIdx.x * blockDim.x + threadIdx.x;
  if (i < NN * DD) acc[i] = 0.0f;
  if (i < NN) { mx[i] = -INFINITY; sm[i] = 0.0f; }
}

// ---------------------------------------------------------------------------
// Pass 1: score_e = att . leaky_relu(xl[src] + xr[dst]); segment max via atomics.
// One wave per edge; float4 feature loads; shuffle reduction.
// Edges e >= EE are the implicit self loops (src = dst = e - EE).
// ---------------------------------------------------------------------------
__global__ void __launch_bounds__(256) edge_score(
    const float* __restrict__ xl, const float* __restrict__ xr,
    const int* __restrict__ src, const int* __restrict__ dst,
    const float* __restrict__ att, float* __restrict__ sc,
    float* __restrict__ mx) {
  const int e = blockIdx.x * (blockDim.x >> 5) + (threadIdx.x >> 5);
  if (e >= ETOT) return;
  const int lane = threadIdx.x & 31;
  const int s = (e < EE) ? src[e] : (e - EE);
  const int d = (e < EE) ? dst[e] : (e - EE);

  const float4* a4 = (const float4*)(xl + (size_t)s * DD);
  const float4* b4 = (const float4*)(xr + (size_t)d * DD);
  const float4* t4 = (const float4*)att;
  float acc = 0.0f;
#pragma unroll
  for (int i = 0; i < 2; ++i) {
    const int idx = lane + 32 * i;              // 64 float4 per 256-wide row
    float4 av = a4[idx], bv = b4[idx], tv = t4[idx];
    acc += lrelu(av.x + bv.x) * tv.x;
    acc += lrelu(av.y + bv.y) * tv.y;
    acc += lrelu(av.z + bv.z) * tv.z;
    acc += lrelu(av.w + bv.w) * tv.w;
  }
#pragma unroll
  for (int off = 16; off > 0; off >>= 1) acc += __shfl_xor(acc, off, 32);
  if (lane == 0) {
    sc[e] = acc;
    atomicMaxF(&mx[d], acc);
  }
}

// ---------------------------------------------------------------------------
// Pass 2: e_e = exp(score_e - max[dst]); segment sum via atomics.
// ---------------------------------------------------------------------------
__global__ void edge_exp(const int* __restrict__ dst, const float* __restrict__ mx,
                         float* __restrict__ sc, float* __restrict__ sm) {
  int e = blockIdx.x * blockDim.x + threadIdx.x;
  if (e >= ETOT) return;
  int d = (e < EE) ? dst[e] : (e - EE);
  float ev = expf(sc[e] - mx[d]);
  sc[e] = ev;
  atomicAdd(&sm[d], ev);
}

// ---------------------------------------------------------------------------
// Pass 3: acc[dst] += (e_e / sum[dst]) * xl[src].  Wave per edge, 8 ch/lane.
// ---------------------------------------------------------------------------
__global__ void __launch_bounds__(256) edge_scatter(
    const float* __restrict__ xl, const int* __restrict__ src,
    const int* __restrict__ dst, const float* __restrict__ sc,
    const float* __restrict__ sm, float* __restrict__ acc) {
  const int e = blockIdx.x * (blockDim.x >> 5) + (threadIdx.x >> 5);
  if (e >= ETOT) return;
  const int lane = threadIdx.x & 31;
  const int s = (e < EE) ? src[e] : (e - EE);
  const int d = (e < EE) ? dst[e] : (e - EE);
  const float alpha = sc[e] / sm[d];

  const float4* a4  = (const float4*)(xl + (size_t)s * DD);
  float*        out = acc + (size_t)d * DD;
#pragma unroll
  for (int i = 0; i < 2; ++i) {
    const int idx = lane + 32 * i;
    float4 v = a4[idx];
    atomicAdd(out + idx * 4 + 0, alpha * v.x);
    atomicAdd(out + idx * 4 + 1, alpha * v.y);
    atomicAdd(out + idx * 4 + 2, alpha * v.z);
    atomicAdd(out + idx * 4 + 3, alpha * v.w);
  }
}

// h = relu(acc + bias)
__global__ void bias_relu(const float* __restrict__ acc,
                          const float* __restrict__ bias,
                          float* __restrict__ h) {
  int i = blockIdx.x * blockDim.x + threadIdx.x;
  if (i < NN * DD) h[i] = fmaxf(acc[i] + bias[i & (DD - 1)], 0.0f);
}

// Row-wise log_softmax over 64 logits: one wave per row, 2 values per lane.
__global__ void log_softmax_rows(const float* __restrict__ logits,
                                 float* __restrict__ out) {
  const int row = blockIdx.x;
  const int lane = threadIdx.x;
  float a = logits[row * DOUT + lane];
  float b = logits[row * DOUT + 32 + lane];
  float m = fmaxf(a, b);
#pragma unroll
  for (int off = 16; off > 0; off >>= 1) m = fmaxf(m, __shfl_xor(m, off, 32));
  float se = expf(a - m) + expf(b - m);
#pragma unroll
  for (int off = 16; off > 0; off >>= 1) se += __shfl_xor(se, off, 32);
  float lse = m + logf(se);
  out[row * DOUT + lane] = a - lse;
  out[row * DOUT + 32 + lane] = b - lse;
}

// ---------------------------------------------------------------------------
extern "C" void kernel_launch(void* const* d_in, const int* in_sizes, int n_in,
                              void* d_out, int out_size, void* d_ws, size_t ws_size,
                              hipStream_t stream) {
  const float* x     = (const float*)d_in[0];
  const float* W1l   = (const float*)d_in[1];
  const float* b1l   = (const float*)d_in[2];
  const float* W1r   = (const float*)d_in[3];
  const float* b1r   = (const float*)d_in[4];
  const float* att1  = (const float*)d_in[5];
  const float* bias1 = (const float*)d_in[6];
  const float* W2l   = (const float*)d_in[7];
  const float* b2l   = (const float*)d_in[8];
  const float* W2r   = (const float*)d_in[9];
  const float* b2r   = (const float*)d_in[10];
  const float* att2  = (const float*)d_in[11];
  const float* bias2 = (const float*)d_in[12];
  const float* Wout  = (const float*)d_in[13];
  const float* bout  = (const float*)d_in[14];
  const int*   eidx  = (const int*)d_in[15];   // edge_index flattened [2, E]
  const int* srcI = eidx;
  const int* dstI = eidx + EE;

  // Workspace layout (~42.5 MB of f32).
  float* ws  = (float*)d_ws;
  float* xl  = ws;  ws += (size_t)NN * DD;
  float* xr  = ws;  ws += (size_t)NN * DD;
  float* acc = ws;  ws += (size_t)NN * DD;
  float* h   = ws;  ws += (size_t)NN * DD;
  float* sc  = ws;  ws += ETOT;
  float* mx  = ws;  ws += NN;
  float* sm  = ws;  ws += NN;

  const int  mTiles = (NN + 31) / 32;          // 313 blocks of 32 rows
  const dim3 gGemm(mTiles, 2);                 // 8 waves cover 128 cols each
  const dim3 bGemm(256);
  const int edgeBlocks = (ETOT + 7) / 8;       // 8 waves (edges) per block
  const int flatBlocks = (NN * DD + 255) / 256;
  const int expBlocks  = (ETOT + 255) / 256;

  // ---- Layer 1 ----
  gemm_wmma<DD><<<gGemm, bGemm, 0, stream>>>(x, W1l, b1l, xl, NN);
  gemm_wmma<DD><<<gGemm, bGemm, 0, stream>>>(x, W1r, b1r, xr, NN);
  init_accum<<<flatBlocks, 256, 0, stream>>>(acc, mx, sm);
  edge_score<<<edgeBlocks, 256, 0, stream>>>(xl, xr, srcI, dstI, att1, sc, mx);
  edge_exp<<<expBlocks, 256, 0, stream>>>(dstI, mx, sc, sm);
  edge_scatter<<<edgeBlocks, 256, 0, stream>>>(xl, srcI, dstI, sc, sm, acc);
  bias_relu<<<flatBlocks, 256, 0, stream>>>(acc, bias1, h);

  // ---- Layer 2 ----
  gemm_wmma<DD><<<gGemm, bGemm, 0, stream>>>(h, W2l, b2l, xl, NN);
  gemm_wmma<DD><<<gGemm, bGemm, 0, stream>>>(h, W2r, b2r, xr, NN);
  init_accum<<<flatBlocks, 256, 0, stream>>>(acc, mx, sm);
  edge_score<<<edgeBlocks, 256, 0, stream>>>(xl, xr, srcI, dstI, att2, sc, mx);
  edge_exp<<<expBlocks, 256, 0, stream>>>(dstI, mx, sc, sm);
  edge_scatter<<<edgeBlocks, 256, 0, stream>>>(xl, srcI, dstI, sc, sm, acc);
  bias_relu<<<flatBlocks, 256, 0, stream>>>(acc, bias2, h);

  // ---- Output head: logits = h @ Wout + bout, then log_softmax ----
  gemm_wmma<DOUT><<<dim3(mTiles, 1), dim3(128), 0, stream>>>(h, Wout, bout, xl, NN);
  log_softmax_rows<<<NN, 32, 0, stream>>>(xl, (float*)d_out);
}